// MultiModelMLP_44152263803448
// MI455X (gfx1250) — compile-verified
//
#include <hip/hip_runtime.h>
#include <math.h>
#include <stdint.h>

#define NM   64
#define INF  6
#define HID  64
#define OUTF 3
#define TWO_PI 6.283185307179586f

typedef __attribute__((ext_vector_type(2))) float v2f;
typedef __attribute__((ext_vector_type(8))) float v8f;
typedef __attribute__((ext_vector_type(4))) int   v4i;

#if defined(__HIP_DEVICE_COMPILE__) && __has_builtin(__builtin_amdgcn_global_load_async_to_lds_b128)
#define HAVE_ASYNC_LDS 1
typedef __attribute__((address_space(1))) v4i* as1_v4i;
typedef __attribute__((address_space(3))) v4i* as3_v4i;
#endif

__device__ __forceinline__ v8f wmma4(v2f a, v2f b, v8f c) {
    // D(16x16,f32) = A(16x4,f32) x B(4x16,f32) + C
    return __builtin_amdgcn_wmma_f32_16x16x4_f32(
        /*neg_a=*/false, a, /*neg_b=*/false, b,
        /*c_mod=*/(short)0, c, /*reuse_a=*/false, /*reuse_b=*/false);
}

// ---------------- kernel 1: zero counters ----------------
__global__ void init_kernel(int* counts, int* cursors) {
    int i = threadIdx.x;
    if (i < NM) { counts[i] = 0; cursors[i] = 0; }
}

// ---------------- kernel 2: routing index + histogram + index output ----------------
__global__ void index_kernel(const float* __restrict__ inputs,
                             float* __restrict__ out_idx,
                             int* __restrict__ widx,
                             int* __restrict__ counts, int B) {
    int i = blockIdx.x * blockDim.x + threadIdx.x;
    if (i >= B) return;
    float x0 = inputs[(size_t)i * INF + 0];
    float x2 = inputs[(size_t)i * INF + 2];
    float ang = atan2f(x2, x0);                       // (-pi, pi]
    float t = fmodf(ang + TWO_PI, TWO_PI) * ((float)NM / TWO_PI);
    int k = (int)floorf(t);
    k = k < 0 ? 0 : (k > NM - 1 ? NM - 1 : k);
    widx[i] = k;
    out_idx[i] = (float)k;
    atomicAdd(&counts[k], 1);
}

// ---------------- kernel 3: exclusive scan over 64 counts ----------------
__global__ void scan_kernel(const int* __restrict__ counts, int* __restrict__ offsets) {
    int acc = 0;
    for (int i = 0; i < NM; ++i) { offsets[i] = acc; acc += counts[i]; }
}

// ---------------- kernel 4: scatter sample ids into per-model buckets ----------------
__global__ void scatter_kernel(const int* __restrict__ widx,
                               const int* __restrict__ offsets,
                               int* __restrict__ cursors,
                               int* __restrict__ perm, int B) {
    int i = blockIdx.x * blockDim.x + threadIdx.x;
    if (i >= B) return;
    int k = widx[i];
    int pos = atomicAdd(&cursors[k], 1);
    perm[offsets[k] + pos] = i;
}

// ---------------- kernel 5: fill constant logits(1.0) + probs(1/64), float4 wide ----------------
__global__ void fill_kernel(float4* __restrict__ p, size_t n_ones, size_t n_tot) {
    size_t i = (size_t)blockIdx.x * blockDim.x + threadIdx.x;
    size_t stride = (size_t)gridDim.x * blockDim.x;
    for (; i < n_tot; i += stride) {
        float v = (i < n_ones) ? 1.0f : 0.015625f;
        p[i] = make_float4(v, v, v, v);
    }
}

// ---------------- kernel 6: grouped WMMA MLP ----------------
#define XPITCH 68   // 64 + 4 pad -> conflict-free LDS (stride 68 dwords, 8B aligned rows)
#define W0P    12   // w0 row pitch: 12*L mod 64 distinct for L=0..15, rows 8B aligned

__global__ __launch_bounds__(128)
void mlp_kernel(const float* __restrict__ inputs,
                const float* __restrict__ w0, const float* __restrict__ b0,
                const float* __restrict__ w1, const float* __restrict__ b1,
                const float* __restrict__ w2, const float* __restrict__ b2,
                const float* __restrict__ w3, const float* __restrict__ b3,
                const float* __restrict__ w4, const float* __restrict__ b4,
                const int* __restrict__ counts, const int* __restrict__ offsets,
                const int* __restrict__ perm,
                float* __restrict__ out_model, float* __restrict__ out_top) {
    __shared__ float wlds[3][HID * XPITCH];     // staged w1..w3: 52224 B
    __shared__ float w0p[HID * W0P];            // w0, K padded 6->8: 3072 B
    __shared__ float w4p[16 * XPITCH];          // w4, N padded 3->16: 4352 B
    __shared__ float blds[4 * HID + 4];         // b0..b3 + b4: 1040 B
    __shared__ float xbuf[4][2][16 * XPITCH];   // per-wave double-buffered tiles: 34816 B

    const int m    = blockIdx.x;               // model id
    const int wave = threadIdx.x >> 5;
    const int lane = threadIdx.x & 31;
    const int row16 = lane & 15;
    const int hi    = lane >> 4;               // 0: K pair {4k,4k+1}, 1: {4k+2,4k+3}
    const int rr    = hi * 8;                  // D-row base for this half-wave

    // ---- block prologue: stage ALL per-model parameters into LDS
    {
        const float* wsrc[3] = { w1 + (size_t)m * HID * HID,
                                 w2 + (size_t)m * HID * HID,
                                 w3 + (size_t)m * HID * HID };
        for (int l = 0; l < 3; ++l) {
            const float4* src = (const float4*)wsrc[l];
            for (int idx = threadIdx.x; idx < HID * (HID / 4); idx += 128) {
                const int o = idx >> 4;        // row (output feature)
                const int q = idx & 15;        // float4 within row
                float* dst = &wlds[l][o * XPITCH + q * 4];
#if defined(HAVE_ASYNC_LDS)
                __builtin_amdgcn_global_load_async_to_lds_b128(
                    (as1_v4i)(src + idx), (as3_v4i)dst, 0, 0);
#else
                *(float4*)dst = src[idx];
#endif
            }
        }
        const float* g0 = w0 + (size_t)m * HID * INF;
        for (int idx = threadIdx.x; idx < HID * 8; idx += 128) {
            const int o = idx >> 3, i = idx & 7;
            w0p[o * W0P + i] = (i < INF) ? g0[o * INF + i] : 0.f;
        }
        const float* g4 = w4 + (size_t)m * OUTF * HID;
        for (int idx = threadIdx.x; idx < 16 * HID; idx += 128) {
            const int o = idx >> 6, i = idx & 63;
            w4p[o * XPITCH + i] = (o < OUTF) ? g4[o * HID + i] : 0.f;
        }
        for (int idx = threadIdx.x; idx < 4 * HID; idx += 128) {
            const int l = idx >> 6, i = idx & 63;
            const float* gb = (l == 0 ? b0 : l == 1 ? b1 : l == 2 ? b2 : b3) + (size_t)m * HID;
            blds[idx] = gb[i];
        }
        if (threadIdx.x < OUTF) blds[4 * HID + threadIdx.x] = b4[(size_t)m * OUTF + threadIdx.x];
#if defined(HAVE_ASYNC_LDS)
        asm volatile("s_wait_asynccnt 0" ::: "memory");
#endif
    }
    __syncthreads();

    const int cnt = counts[m];
    const int off = offsets[m];
    const int ntiles = (cnt + 15) >> 4;
    const int slot = blockIdx.y * 4 + wave;    // 0..63 persistent wave slot

    for (int t = slot; t < ntiles; t += 64) {
        const int base = t * 16;
        float* cur = &xbuf[wave][0][0];
        float* nxt = &xbuf[wave][1][0];

        // ---- stage layer-0 inputs (16 rows x 6 feats, pad K to 8 with zeros)
        if (lane < 16) {
            int r = base + lane; if (r >= cnt) r = cnt - 1;
            int s = perm[off + r];
            const float* ip = inputs + (size_t)s * INF;
            #pragma unroll
            for (int c = 0; c < INF; ++c) cur[lane * XPITCH + c] = ip[c];
            cur[lane * XPITCH + 6] = 0.f;
            cur[lane * XPITCH + 7] = 0.f;
        }
        asm volatile("s_wait_dscnt 0" ::: "memory");

        // ---- layer 0: 6(pad 8) -> 64, ReLU (all operands LDS-resident)
        {
            #pragma unroll
            for (int ct = 0; ct < 4; ++ct) {
                const int ob = ct * 16;
                v8f acc = {};
                #pragma unroll
                for (int k = 0; k < 2; ++k) {
                    const int koff = 4 * k + hi * 2;
                    v2f a  = *(const v2f*)&cur[row16 * XPITCH + koff];
                    v2f bf = *(const v2f*)&w0p[(ob + row16) * W0P + koff];
                    acc = wmma4(a, bf, acc);
                }
                const float bias = blds[ob + row16];
                #pragma unroll
                for (int j = 0; j < 8; ++j) {
                    float v = acc[j] + bias;
                    nxt[(rr + j) * XPITCH + ob + row16] = v > 0.f ? v : 0.f;
                }
            }
            asm volatile("s_wait_dscnt 0" ::: "memory");
            float* tmp = cur; cur = nxt; nxt = tmp;
        }

        // ---- hidden layers 1..3: 64 -> 64, ReLU; A and B fragments from LDS
        for (int l = 0; l < 3; ++l) {
            const float* W = &wlds[l][0];
            #pragma unroll
            for (int ct = 0; ct < 4; ++ct) {
                const int ob = ct * 16;
                v8f acc = {};
                #pragma unroll
                for (int k = 0; k < 16; ++k) {
                    const int koff = 4 * k + hi * 2;
                    v2f a  = *(const v2f*)&cur[row16 * XPITCH + koff];
                    v2f bf = *(const v2f*)&W[(ob + row16) * XPITCH + koff];
                    acc = wmma4(a, bf, acc);
                }
                const float bias = blds[(l + 1) * HID + ob + row16];
                #pragma unroll
                for (int j = 0; j < 8; ++j) {
                    float v = acc[j] + bias;
                    nxt[(rr + j) * XPITCH + ob + row16] = v > 0.f ? v : 0.f;
                }
            }
            asm volatile("s_wait_dscnt 0" ::: "memory");
            float* tmp = cur; cur = nxt; nxt = tmp;
        }

        // ---- layer 4: 64 -> 3 (N padded to 16 with zero rows), write outputs
        {
            v8f acc = {};
            #pragma unroll
            for (int k = 0; k < 16; ++k) {
                const int koff = 4 * k + hi * 2;
                v2f a  = *(const v2f*)&cur[row16 * XPITCH + koff];
                v2f bf = *(const v2f*)&w4p[row16 * XPITCH + koff];
                acc = wmma4(a, bf, acc);
            }
            if (row16 < OUTF) {
                const float bias = blds[4 * HID + row16];
                #pragma unroll
                for (int j = 0; j < 8; ++j) {
                    const int r = base + rr + j;
                    if (r < cnt) {
                        const int s = perm[off + r];
                        const float v = acc[j] + bias;
                        out_model[(size_t)s * OUTF + row16] = v;
                        out_top  [(size_t)s * OUTF + row16] = v;
                    }
                }
            }
        }
    }
}

extern "C" void kernel_launch(void* const* d_in, const int* in_sizes, int n_in,
                              void* d_out, int out_size, void* d_ws, size_t ws_size,
                              hipStream_t stream) {
    (void)n_in; (void)out_size; (void)ws_size;
    const float* inputs = (const float*)d_in[0];
    const float* w0 = (const float*)d_in[1];  const float* b0 = (const float*)d_in[2];
    const float* w1 = (const float*)d_in[3];  const float* b1 = (const float*)d_in[4];
    const float* w2 = (const float*)d_in[5];  const float* b2 = (const float*)d_in[6];
    const float* w3 = (const float*)d_in[7];  const float* b3 = (const float*)d_in[8];
    const float* w4 = (const float*)d_in[9];  const float* b4 = (const float*)d_in[10];

    const int B = in_sizes[0] / INF;

    // output layout: model(3B) | top(3B) | idx(B) | logits(64B) | probs(64B)
    float* out        = (float*)d_out;
    float* out_model  = out;
    float* out_top    = out + (size_t)3 * B;
    float* out_idx    = out + (size_t)6 * B;
    float* out_lp     = out + (size_t)7 * B;      // logits then probs, contiguous

    // workspace layout
    int* counts  = (int*)d_ws;
    int* offsets = counts + NM;
    int* cursors = offsets + NM;
    int* widx    = cursors + NM;
    int* perm    = widx + B;

    init_kernel<<<1, 64, 0, stream>>>(counts, cursors);
    index_kernel<<<(B + 255) / 256, 256, 0, stream>>>(inputs, out_idx, widx, counts, B);
    scan_kernel<<<1, 1, 0, stream>>>(counts, offsets);
    scatter_kernel<<<(B + 255) / 256, 256, 0, stream>>>(widx, offsets, cursors, perm, B);

    const size_t n_ones = (size_t)NM * B / 4;     // logits region in float4 units
    const size_t n_tot  = (size_t)2 * NM * B / 4; // logits + probs
    fill_kernel<<<4096, 256, 0, stream>>>((float4*)out_lp, n_ones, n_tot);

    mlp_kernel<<<dim3(NM, 16), 128, 0, stream>>>(
        inputs, w0, b0, w1, b1, w2, b2, w3, b3, w4, b4,
        counts, offsets, perm, out_model, out_top);
}